// SWN_89172110999963
// MI455X (gfx1250) — compile-verified
//
#include <hip/hip_runtime.h>

// Flow-warp (bilinear grid sample) for im=[8,256,256,64] f32, flow=[8,256,256,2] f32.
// Memory-bound gather: ~272 MB traffic -> ~12us floor at 23.3 TB/s on MI455X.
// Uses gfx1250 async load-to-LDS to stage flow, B128 vector gathers, NT store.

#define HH 256
#define WW 256
#define CC 64
#define PIX_PER_BLK 16
#define THREADS 256   // 16 pixels * 16 float4 chunks

typedef float float4v __attribute__((ext_vector_type(4)));
typedef float float2v __attribute__((ext_vector_type(2)));

// Types matching the async-to-LDS builtin signature discovered via compile probe:
// param0: v2i32 in AS1 (global), param1: v2i32 in AS3 (LDS), then (imm, imm).
typedef int v2i __attribute__((vector_size(2 * sizeof(int))));
typedef __attribute__((address_space(1))) v2i* gv2i_p;
typedef __attribute__((address_space(3))) v2i* lv2i_p;

struct PixParam {
  int x0, x1, y0, y1;
  float wa, wb, wc, wd;
};

__device__ __forceinline__ int clampi(int v, int lo, int hi) {
  return v < lo ? lo : (v > hi ? hi : v);
}

__global__ __launch_bounds__(THREADS) void
flow_warp_kernel(const float* __restrict__ im,
                 const float* __restrict__ flow,
                 float* __restrict__ out) {
  __shared__ float    s_flow[PIX_PER_BLK * 2];
  __shared__ PixParam s_p[PIX_PER_BLK];

  const int      tid     = threadIdx.x;
  const unsigned basePix = (unsigned)blockIdx.x * PIX_PER_BLK; // < 2^23
  const unsigned b       = basePix >> 16;          // H*W == 65536
  const unsigned irow    = (basePix >> 8) & 255u;  // row (blocks never cross rows)
  const unsigned j0      = basePix & 255u;

  // ---- Phase 1: wave-0 lanes 0..15 stage flow into LDS and compute params ----
  if (tid < PIX_PER_BLK) {
#if __has_builtin(__builtin_amdgcn_global_load_async_to_lds_b64)
    // gfx1250 async path: ASYNCcnt-tracked copy of one float2 per lane into LDS.
    __builtin_amdgcn_global_load_async_to_lds_b64(
        (gv2i_p)(flow + (size_t)(basePix + tid) * 2),
        (lv2i_p)&s_flow[tid * 2],
        0, 0);
#if __has_builtin(__builtin_amdgcn_s_wait_asynccnt)
    __builtin_amdgcn_s_wait_asynccnt(0);
#else
    asm volatile("s_wait_asynccnt 0" ::: "memory");
#endif
#else
    // Fallback: plain non-temporal load (flow is streamed once).
    const float2v f2 = __builtin_nontemporal_load(
        reinterpret_cast<const float2v*>(flow) + (basePix + tid));
    s_flow[tid * 2 + 0] = f2.x;
    s_flow[tid * 2 + 1] = f2.y;
#endif
    const float fx = s_flow[tid * 2 + 0];
    const float fy = s_flow[tid * 2 + 1];

    // Reference math (faithful): linspace(-1,1,N) grid, flow/H for both axes,
    // map with *dim/2, floor, clip BOTH x and y to H-1.
    const float gx = -1.0f + (float)(j0 + tid) * (2.0f / (float)(WW - 1));
    const float gy = -1.0f + (float)irow       * (2.0f / (float)(HH - 1));
    float x = gx + fx * (1.0f / (float)HH);
    float y = gy + fy * (1.0f / (float)HH);
    x = (x + 1.0f) * ((float)WW * 0.5f);
    y = (y + 1.0f) * ((float)HH * 0.5f);

    int x0 = (int)floorf(x);
    int y0 = (int)floorf(y);
    int x1 = x0 + 1;
    int y1 = y0 + 1;
    const int mx = HH - 1;
    x0 = clampi(x0, 0, mx); x1 = clampi(x1, 0, mx);
    y0 = clampi(y0, 0, mx); y1 = clampi(y1, 0, mx);

    const float x0f = (float)x0, x1f = (float)x1;
    const float y0f = (float)y0, y1f = (float)y1;

    PixParam p;
    p.x0 = x0; p.x1 = x1; p.y0 = y0; p.y1 = y1;
    p.wa = (x1f - x) * (y1f - y);
    p.wb = (x1f - x) * (y - y0f);
    p.wc = (x - x0f) * (y1f - y);
    p.wd = (x - x0f) * (y - y0f);
    s_p[tid] = p;
  }
  __syncthreads();

  // ---- Phase 2: 16 lanes per pixel, one float4 channel-chunk each ----
  const int p = tid >> 4;   // pixel within block
  const int q = tid & 15;   // float4 chunk within 64 channels
  const PixParam pp = s_p[p];

  const float* imB = im + ((size_t)b * (size_t)(HH * WW * CC));
  const float4v Ia = *((const float4v*)(imB + ((size_t)(pp.y0 * WW + pp.x0) * CC)) + q);
  const float4v Ib = *((const float4v*)(imB + ((size_t)(pp.y1 * WW + pp.x0) * CC)) + q);
  const float4v Ic = *((const float4v*)(imB + ((size_t)(pp.y0 * WW + pp.x1) * CC)) + q);
  const float4v Id = *((const float4v*)(imB + ((size_t)(pp.y1 * WW + pp.x1) * CC)) + q);

  float4v o = pp.wa * Ia;
  o = o + pp.wb * Ib;
  o = o + pp.wc * Ic;
  o = o + pp.wd * Id;

  // Output is write-once streaming: non-temporal so it doesn't evict im from L2.
  __builtin_nontemporal_store(
      o, (float4v*)out + ((size_t)(basePix + p) * (CC / 4) + q));
}

extern "C" void kernel_launch(void* const* d_in, const int* in_sizes, int n_in,
                              void* d_out, int out_size, void* d_ws, size_t ws_size,
                              hipStream_t stream) {
  const float* im   = (const float*)d_in[0];
  const float* flow = (const float*)d_in[1];
  float*       out  = (float*)d_out;

  const int pixels = in_sizes[1] / 2;           // B*H*W
  dim3 grid((unsigned)(pixels / PIX_PER_BLK));
  dim3 block(THREADS);
  hipLaunchKernelGGL(flow_warp_kernel, grid, block, 0, stream, im, flow, out);
}